// RandCropResize_39822936768612
// MI455X (gfx1250) — compile-verified
//
#include <hip/hip_runtime.h>

#define NB 64
#define NC 3
#define NH 512
#define NW 512

typedef float v4f __attribute__((ext_vector_type(4)));
typedef int   v4i __attribute__((ext_vector_type(4)));
typedef v4i __attribute__((address_space(1))) v4i_glob;   // global (AS1) int4
typedef v4i __attribute__((address_space(3))) v4i_lds;    // LDS (AS3) int4

#if defined(__AMDGCN__) && __has_builtin(__builtin_amdgcn_global_load_async_to_lds_b128) && \
    __has_builtin(__builtin_amdgcn_s_wait_asynccnt)
#define USE_ASYNC_LDS 1
#else
#define USE_ASYNC_LDS 0
#endif

__global__ __launch_bounds__(128) void rand_crop_resize_kernel(
    const float* __restrict__ img,
    const int* __restrict__ y1v, const int* __restrict__ y2v,
    const int* __restrict__ x1v, const int* __restrict__ x2v,
    float* __restrict__ out)
{
    // 6 staged input rows: [channel*2 + {iy0,iy1}][x]
    __shared__ float lds[2 * NC][NW];

    const int t   = threadIdx.x;          // 0..127, 4 px each
    const int row = blockIdx.x;           // b*NH + y
    const int b   = row >> 9;             // / NH
    const int y   = row & (NH - 1);

    const int y1 = y1v[b], y2 = y2v[b], x1 = x1v[b], x2 = x2v[b];
    const int ny = y2 - y1;
    const int nx = x2 - x1;

    // ---- vertical source coords (uniform across the block) ----
    float sy = ((float)y + 0.5f) * (float)ny * (1.0f / 512.0f) - 0.5f;
    sy = fminf(fmaxf(sy, 0.0f), (float)(ny - 1));
    const int   j0 = (int)sy;             // sy >= 0, floor == trunc
    const float wy = sy - (float)j0;
    const int   j1 = (j0 + 1 < ny - 1) ? (j0 + 1) : (ny - 1);
    const int   iy0 = y1 + j0;
    const int   iy1 = y1 + j1;

    // ---- stage 6 rows (2 per channel) into LDS, 16B per lane per row ----
    const float* base = img + (size_t)b * (NC * NH * NW);
#pragma unroll
    for (int c = 0; c < NC; ++c) {
        const float* r0 = base + ((size_t)c * NH + (size_t)iy0) * NW + t * 4;
        const float* r1 = base + ((size_t)c * NH + (size_t)iy1) * NW + t * 4;
#if USE_ASYNC_LDS
        __builtin_amdgcn_global_load_async_to_lds_b128(
            (v4i_glob*)r0, (v4i_lds*)&lds[2 * c + 0][t * 4], 0, 0);
        __builtin_amdgcn_global_load_async_to_lds_b128(
            (v4i_glob*)r1, (v4i_lds*)&lds[2 * c + 1][t * 4], 0, 0);
#else
        *(v4f*)&lds[2 * c + 0][t * 4] = *(const v4f*)r0;
        *(v4f*)&lds[2 * c + 1][t * 4] = *(const v4f*)r1;
#endif
    }
#if USE_ASYNC_LDS
    __builtin_amdgcn_s_wait_asynccnt(0);
#endif
    __syncthreads();

    // ---- horizontal source coords for this thread's 4 output pixels ----
    int   xa0[4], xa1[4];
    float wx[4];
#pragma unroll
    for (int k = 0; k < 4; ++k) {
        const int x = t * 4 + k;
        float sx = ((float)x + 0.5f) * (float)nx * (1.0f / 512.0f) - 0.5f;
        sx = fminf(fmaxf(sx, 0.0f), (float)(nx - 1));
        const int i0 = (int)sx;
        wx[k]  = sx - (float)i0;
        const int i1 = (i0 + 1 < nx - 1) ? (i0 + 1) : (nx - 1);
        xa0[k] = x1 + i0;
        xa1[k] = x1 + i1;
    }

    // ---- bilinear blend from LDS, one NT b128 store per channel ----
    float* outbase = out + (size_t)b * (NC * NH * NW) + (size_t)y * NW + t * 4;
#pragma unroll
    for (int c = 0; c < NC; ++c) {
        v4f v;
#pragma unroll
        for (int k = 0; k < 4; ++k) {
            const float a  = lds[2 * c + 0][xa0[k]];
            const float bb = lds[2 * c + 0][xa1[k]];
            const float cc = lds[2 * c + 1][xa0[k]];
            const float dd = lds[2 * c + 1][xa1[k]];
            const float top = a  + (bb - a)  * wx[k];
            const float bot = cc + (dd - cc) * wx[k];
            v[k] = top + (bot - top) * wy;
        }
        __builtin_nontemporal_store(v, (v4f*)(outbase + (size_t)c * (NH * NW)));
    }
}

extern "C" void kernel_launch(void* const* d_in, const int* in_sizes, int n_in,
                              void* d_out, int out_size, void* d_ws, size_t ws_size,
                              hipStream_t stream) {
    (void)in_sizes; (void)n_in; (void)out_size; (void)d_ws; (void)ws_size;
    const float* img = (const float*)d_in[0];
    const int*   y1  = (const int*)d_in[1];
    const int*   y2  = (const int*)d_in[2];
    const int*   x1  = (const int*)d_in[3];
    const int*   x2  = (const int*)d_in[4];
    float*       out = (float*)d_out;

    dim3 grid(NB * NH);   // one block per (batch, output row)
    dim3 block(128);      // 4 waves of 32; each thread -> 4 px x 3 channels
    rand_crop_resize_kernel<<<grid, block, 0, stream>>>(img, y1, y2, x1, x2, out);
}